// CRF_4355096838905
// MI455X (gfx1250) — compile-verified
//
#include <hip/hip_runtime.h>
#include <stdint.h>

// Problem constants (B, S, T) = (128, 512, 64)
#define BB 128
#define SS 512
#define TTAG 64
#define START_TAG 62
#define STOP_TAG 63
#define PF 4  // async prefetch depth (power of 2)

// ---------- CDNA5 async global->LDS plumbing ----------
#if __has_builtin(__builtin_amdgcn_s_wait_asynccnt)
#define WAIT_ASYNC(n) __builtin_amdgcn_s_wait_asynccnt(n)
#else
#define WAIT_ASYNC(n) asm volatile("s_wait_asynccnt %0" ::"i"(n) : "memory")
#endif

typedef __attribute__((address_space(1))) int as1_int;  // global
typedef __attribute__((address_space(3))) int as3_int;  // LDS

typedef unsigned int u32x4 __attribute__((ext_vector_type(4)));
typedef int i32x8 __attribute__((ext_vector_type(8)));
typedef int i32x4 __attribute__((ext_vector_type(4)));

__device__ __forceinline__ void async_load_f32_to_lds(const float* gptr, float* lptr) {
#if __has_builtin(__builtin_amdgcn_global_load_async_to_lds_b32)
  __builtin_amdgcn_global_load_async_to_lds_b32(
      (as1_int*)gptr, (as3_int*)lptr, /*offset=*/0, /*cpol=*/0);
#else
  uint32_t lds_off = (uint32_t)(uintptr_t)(as3_int*)lptr;
  uint64_t gaddr = (uint64_t)(uintptr_t)gptr;
  asm volatile("global_load_async_to_lds_b32 %0, %1, off"
               ::"v"(lds_off), "v"(gaddr)
               : "memory");
#endif
}

// ---------- Viterbi decode kernel: 1 block / batch, lane = tag j ----------
__global__ __launch_bounds__(TTAG) void crf_viterbi_kernel(
    const float* __restrict__ feats,   // (B, S, T)
    const int* __restrict__ mask,      // (B, S) 0/1
    const float* __restrict__ trans,   // (T, T)  [i=prev][j=cur]
    float* __restrict__ out)           // (B, S) decoded tags
{
  __shared__ alignas(16) float   transLDS[TTAG * TTAG];  // 16 KB, [i*64 + j]
  __shared__ alignas(16) uint8_t bpLDS[SS * TTAG];       // 32 KB backpointers
  __shared__ alignas(16) float   featsBuf[PF][TTAG];     // async prefetch ring
  __shared__ alignas(16) float   partBuf[2][TTAG];       // double-buffered partition
  __shared__ alignas(16) float   lastPart[TTAG];
  __shared__ alignas(16) int     decodeLDS[SS];
  __shared__ int lenLDS;
  __shared__ int p0LDS;

  const int j = threadIdx.x;  // tag index 0..63
  const int b = blockIdx.x;   // batch index

  // ---- Transition matrix -> LDS. Prefer a single TDM 2D tile DMA (64x64,
  // 4B elements), issued by wave 0 and overlapped with the setup work below;
  // otherwise fall back to a coalesced per-lane copy loop.
#if __has_builtin(__builtin_amdgcn_tensor_load_to_lds) && \
    __has_builtin(__builtin_amdgcn_s_wait_tensorcnt)
#define USE_TDM 1
#else
#define USE_TDM 0
#endif

#if USE_TDM
  if (j < 32) {  // wave 0 only (TDM ignores EXEC; scalar skip for wave 1)
    uint32_t lds_addr = (uint32_t)(uintptr_t)(as3_int*)&transLDS[0];
    uint64_t gaddr = (uint64_t)(uintptr_t)trans;
    // D# group0: count=1 | lds_addr | global_addr[56:0] | type=2
    u32x4 g0;
    g0.x = 1u;                                   // count=1, is_restore=0
    g0.y = lds_addr;                             // lds_addr [63:32]
    g0.z = (uint32_t)gaddr;                      // global_addr low
    g0.w = (uint32_t)(gaddr >> 32) | (2u << 30); // global_addr hi | type=2
    // D# group1: data_size=4B, tensor 64x64, tile 64x64, dim0_stride=64
    i32x8 g1;
    g1[0] = 0x00020000;        // workgroup_mask=0, data_size=2 (4B)
    g1[1] = (TTAG & 0xFFFF) << 16;  // tensor_dim0[15:0] @ bits[63:48]
    g1[2] = (TTAG & 0xFFFF) << 16;  // dim0 hi=0, tensor_dim1[15:0] @ bits[95:80]
    g1[3] = (TTAG & 0xFFFF) << 16;  // dim1 hi=0, tile_dim0 @ bits[127:112]
    g1[4] = TTAG;              // tile_dim1 @ [143:128]; tile_dim2=0
    g1[5] = TTAG;              // tensor_dim0_stride low 32 @ [191:160]
    g1[6] = 0;                 // stride hi, tensor_dim1_stride low
    g1[7] = 0;
    i32x4 gz4 = {0, 0, 0, 0};              // groups 2/3 unused for 2D tensor
    i32x8 gz8 = {0, 0, 0, 0, 0, 0, 0, 0};  // extra group (6-arg toolchain form)
    __builtin_amdgcn_tensor_load_to_lds(g0, g1, gz4, gz4, gz8, /*cpol=*/0);
  }
#else
  for (int k = j; k < TTAG * TTAG; k += TTAG) transLDS[k] = trans[k];
#endif

  // Zero backpointer table (rows >= len must be zero to match reference).
  // Overlaps with the TDM transfer above.
  {
    int* bp32 = (int*)bpLDS;
    for (int k = j; k < SS * TTAG / 4; k += TTAG) bp32[k] = 0;
  }

  // Sequence length from prefix mask.
  {
    const int* mrow = mask + (size_t)b * SS;
    int partial = 0;
    for (int k = j; k < SS; k += TTAG) partial += (mrow[k] != 0) ? 1 : 0;
    decodeLDS[j] = partial;  // reuse as scratch
    __syncthreads();
    if (j == 0) {
      int tot = 0;
      for (int k = 0; k < TTAG; ++k) tot += decodeLDS[k];
      lenLDS = tot;
    }
  }

#if USE_TDM
  if (j < 32) __builtin_amdgcn_s_wait_tensorcnt(0);  // wave 0's DMA done
#endif
  __syncthreads();  // transLDS + lenLDS visible to both waves

  const int len = lenLDS;
  const int last_pos = len - 1;

  const float* frow = feats + (size_t)b * SS * TTAG;

  // Step 0: part0[j] = feats[b,0,j] + trans[START_TAG, j]
  {
    float p0 = frow[j] + transLDS[START_TAG * TTAG + j];
    partBuf[0][j] = p0;
    if (last_pos == 0) lastPart[j] = p0;
  }

  // Preload rows 1..PF-1 asynchronously (each lane fetches its own column j).
  for (int r = 1; r < PF; ++r) {
    async_load_f32_to_lds(frow + (size_t)r * TTAG + j, &featsBuf[r & (PF - 1)][j]);
  }

  // ---------------- forward Viterbi scan ----------------
  int cur = 0;
  for (int s = 1; s <= last_pos; ++s) {
    // Issue prefetch for row s+PF-1 into its logical slot. Clamp the address
    // (stays in-bounds; tail data unused) so the issue count is uniform and
    // s_wait_asynccnt(PF-1) + in-order completion guarantees row s is ready.
    {
      int pr = s + PF - 1;
      int slot = pr & (PF - 1);
      if (pr > SS - 1) pr = SS - 1;
      async_load_f32_to_lds(frow + (size_t)pr * TTAG + j, &featsBuf[slot][j]);
    }
    WAIT_ASYNC(PF - 1);
    asm volatile("" ::: "memory");

    const float fj = featsBuf[s & (PF - 1)][j];
    const float* pc = partBuf[cur];

    // cur[i,j] = (feats[j] + trans[i,j]) + part[i]; first-max argmax (strict >)
    float best = (fj + transLDS[j]) + pc[0];
    int bestI = 0;
#pragma unroll 4
    for (int i = 1; i < TTAG; ++i) {
      float v = (fj + transLDS[i * TTAG + j]) + pc[i];
      if (v > best) { best = v; bestI = i; }
    }

    bpLDS[s * TTAG + j] = (uint8_t)bestI;  // mask true for s < len
    partBuf[cur ^ 1][j] = best;
    if (s == last_pos) lastPart[j] = best;
    cur ^= 1;
    __syncthreads();
  }
  __syncthreads();

  // ---------------- terminal transition ----------------
  // last_values[i,j] = last_partition[i] + trans[i,j]; pointer0 = argmax_i at j=STOP
  {
    float best = lastPart[0] + transLDS[j];
    int bestI = 0;
#pragma unroll 4
    for (int i = 1; i < TTAG; ++i) {
      float v = lastPart[i] + transLDS[i * TTAG + j];
      if (v > best) { best = v; bestI = i; }
    }
    if (j == STOP_TAG) p0LDS = bestI;
  }
  __syncthreads();

  // ---------------- backtrace (sequential chain, all in LDS) ----------------
  if (j == 0) {
    const int p0 = p0LDS;
    int ptr = p0;
    decodeLDS[SS - 1] = ptr;
    for (int t = SS - 2; t >= 0; --t) {
      if (t == last_pos) ptr = p0;                       // at_last override
      else               ptr = (int)bpLDS[(t + 1) * TTAG + ptr];
      decodeLDS[t] = ptr;
    }
  }
  __syncthreads();

  // Coalesced output write (reference output dtype -> float* per harness).
  float* orow = out + (size_t)b * SS;
  for (int t = j; t < SS; t += TTAG) orow[t] = (float)decodeLDS[t];
}

// ---------------- host-side launch ----------------
extern "C" void kernel_launch(void* const* d_in, const int* in_sizes, int n_in,
                              void* d_out, int out_size, void* d_ws, size_t ws_size,
                              hipStream_t stream) {
  (void)in_sizes; (void)n_in; (void)out_size; (void)d_ws; (void)ws_size;
  const float* feats = (const float*)d_in[0];   // (B,S,T) f32
  const int*   mask  = (const int*)d_in[1];     // (B,S) int
  /* d_in[2] = tags — unused by Viterbi decode */
  const float* trans = (const float*)d_in[3];   // (T,T) f32
  float* out = (float*)d_out;                   // (B,S)

  crf_viterbi_kernel<<<BB, TTAG, 0, stream>>>(feats, mask, trans, out);
}